// MachampCRFDecoder_79104707657944
// MI455X (gfx1250) — compile-verified
//
#include <hip/hip_runtime.h>
#include <hip/hip_bf16.h>
#include <math.h>

typedef __attribute__((ext_vector_type(2))) float v2f;
typedef __attribute__((ext_vector_type(8))) float v8f;

#define B_ 64
#define T_ 512
#define D_ 1024
#define N_ 21
#define KC 64          // K-chunk
#define AP 68          // A LDS row stride (floats): conflict-free b64 frag reads
#define BP 68          // B LDS row stride (floats)

// ---------------------------------------------------------------------------
// Kernel 1: logits = mlm_out @ W^T + b   via V_WMMA_F32_16X16X4_F32
// 256 blocks x 256 threads (8 waves). One wave owns a 16-row M tile and two
// 16-wide N tiles. A: reg->wave-private-LDS pipeline (coalesced b128 global
// loads). W: double-buffered LDS in K-major layout (1 ds_load_b64 / fragment).
// ---------------------------------------------------------------------------
__global__ __launch_bounds__(256) void gemm_logits_kernel(
    const float* __restrict__ A,      // (B*T, D)
    const float* __restrict__ W,      // (N_, D)
    const float* __restrict__ bias,   // (N_,)
    float* __restrict__ logits)       // (B*T, N_)
{
    __shared__ float Alds[8][16 * AP];      // per-wave A tile (34816 B total)
    __shared__ float Blds[2][32 * BP];      // double-buffered W chunk (17408 B)

    const int tid   = threadIdx.x;
    const int lane  = tid & 31;
    const int wave  = tid >> 5;
    const int m0    = (blockIdx.x * 8 + wave) * 16;   // 2048 M tiles of 16 rows
    const int ln    = lane & 15;
    const int h     = lane >> 4;                      // half-wave id
    const int khalf = h * 2;                          // K offset per WMMA layout

    v8f c0 = {};   // labels 0..15
    v8f c1 = {};   // labels 16..20 (+pad)

    float4 areg[8];
    float4 wreg[2];
    const float* Abase = A + (size_t)m0 * D_;

    auto load_chunk = [&](int k0) {
#pragma unroll
        for (int r = 0; r < 8; ++r) {   // rows 2r+h, 64 floats via 16 lanes x f4
            const float* p = Abase + (size_t)(2 * r + h) * D_ + k0 + 4 * ln;
            areg[r] = *(const float4*)p;
        }
#pragma unroll
        for (int q = 0; q < 2; ++q) {   // 512 float4 units of W chunk / 256 thr
            const int e  = tid + 256 * q;
            const int n  = e >> 4;
            const int kk = (e & 15) * 4;
            if (n < N_) wreg[q] = *(const float4*)(W + (size_t)n * D_ + k0 + kk);
            else        wreg[q] = make_float4(0.f, 0.f, 0.f, 0.f);
        }
    };

    load_chunk(0);

    for (int c = 0; c < D_ / KC; ++c) {
        const int buf = c & 1;

        // ---- stage regs -> LDS ------------------------------------------
        float* aw = &Alds[wave][0];
#pragma unroll
        for (int r = 0; r < 8; ++r)
            *(float4*)&aw[(2 * r + h) * AP + 4 * ln] = areg[r];
#pragma unroll
        for (int q = 0; q < 2; ++q) {
            const int e  = tid + 256 * q;
            const int n  = e >> 4;
            const int kk = (e & 15) * 4;
            *(float4*)&Blds[buf][n * BP + kk] = wreg[q];
        }
        __syncthreads();

        // ---- prefetch next chunk (overlaps WMMA phase) ------------------
        if (c + 1 < D_ / KC) load_chunk((c + 1) * KC);

        // ---- 16 k-steps of WMMA from LDS --------------------------------
        const float* af  = &Alds[wave][ln * AP + khalf];
        const float* bf0 = &Blds[buf][ln * BP + khalf];
        const float* bf1 = &Blds[buf][(16 + ln) * BP + khalf];
#pragma unroll
        for (int kk = 0; kk < KC; kk += 4) {
            v2f a  = *(const v2f*)(af + kk);
            v2f b0 = *(const v2f*)(bf0 + kk);
            v2f b1 = *(const v2f*)(bf1 + kk);
            c0 = __builtin_amdgcn_wmma_f32_16x16x4_f32(false, a, false, b0,
                                                       (short)0, c0, false, false);
            c1 = __builtin_amdgcn_wmma_f32_16x16x4_f32(false, a, false, b1,
                                                       (short)0, c1, false, false);
        }
        // Writes to Blds[buf] recur 2 chunks later (after the next barrier);
        // all reads above are consumed by WMMAs before this wave signals it.
    }

    // ---- store C: VGPR j -> (M = j or 8+j, N = ln) ----------------------
    const int mbase = m0 + (h ? 8 : 0);
    const float bias0 = bias[ln];
    const float bias1 = (ln + 16 < N_) ? bias[ln + 16] : 0.0f;
#pragma unroll
    for (int j = 0; j < 8; ++j) {
        const size_t m = (size_t)(mbase + j);
        logits[m * N_ + ln] = c0[j] + bias0;
        if (ln + 16 < N_)
            logits[m * N_ + ln + 16] = c1[j] + bias1;
    }
}

// ---------------------------------------------------------------------------
// Kernel 2: CRF log-partition (forward scan) + joint score per batch element.
// One wave32 per batch element; lanes 0..20 own one label each.
// ---------------------------------------------------------------------------
__global__ __launch_bounds__(32) void crf_forward_kernel(
    const float* __restrict__ logits,
    const int*   __restrict__ mask,
    const int*   __restrict__ gold,
    const float* __restrict__ trans,
    const float* __restrict__ start_t,
    const float* __restrict__ end_t,
    float* __restrict__ partial)
{
    const int b = blockIdx.x;
    const int j = threadIdx.x;
    __shared__ float alpha_s[N_];

    const float* lg = logits + (size_t)b * T_ * N_;
    const int*   mk = mask + (size_t)b * T_;
    const int*   g  = gold + (size_t)b * T_;

    float alpha = 0.0f;
    float tcol[N_];
    if (j < N_) {
        alpha = start_t[j] + lg[j];
#pragma unroll
        for (int i = 0; i < N_; ++i) tcol[i] = trans[i * N_ + j];
    }

    for (int t = 1; t < T_; ++t) {
        if (j < N_) alpha_s[j] = alpha;
        __syncthreads();
        if (j < N_) {
            float m = -INFINITY;
#pragma unroll
            for (int i = 0; i < N_; ++i) m = fmaxf(m, alpha_s[i] + tcol[i]);
            float s = 0.0f;
#pragma unroll
            for (int i = 0; i < N_; ++i) s += __expf(alpha_s[i] + tcol[i] - m);
            const float newv = m + __logf(s) + lg[(size_t)t * N_ + j];
            if (mk[t]) alpha = newv;
        }
        __syncthreads();
    }

    // logZ = logsumexp_j(alpha_j + end_t[j]) via wave32 shuffle reduction
    float v = (j < N_) ? (alpha + end_t[j]) : -INFINITY;
    float m = v;
    for (int off = 16; off > 0; off >>= 1) m = fmaxf(m, __shfl_down(m, off, 32));
    m = __shfl(m, 0, 32);
    float e = (j < N_) ? __expf(v - m) : 0.0f;
    for (int off = 16; off > 0; off >>= 1) e += __shfl_down(e, off, 32);

    // Joint score: lane-strided partial sums over t
    float sc = 0.0f;
    for (int t = j; t < T_ - 1; t += 32) {
        sc += trans[g[t] * N_ + g[t + 1]] * (mk[t + 1] ? 1.0f : 0.0f);
        sc += lg[(size_t)t * N_ + g[t]] * (mk[t] ? 1.0f : 0.0f);
    }
    for (int off = 16; off > 0; off >>= 1) sc += __shfl_down(sc, off, 32);

    if (j == 0) {
        const float logZ = m + __logf(e);
        int cnt = 0;
        for (int t = 0; t < T_; ++t) cnt += (mk[t] ? 1 : 0);
        const int last_tag = g[cnt - 1];
        float score = start_t[g[0]] + sc + end_t[last_tag]
                    + lg[(size_t)(T_ - 1) * N_ + last_tag] * (mk[T_ - 1] ? 1.0f : 0.0f);
        partial[b] = score - logZ;
    }
}

// ---------------------------------------------------------------------------
// Kernel 3: Viterbi decode per batch element. Backpointers in LDS (u8).
// ---------------------------------------------------------------------------
__global__ __launch_bounds__(32) void crf_viterbi_kernel(
    const float* __restrict__ logits,
    const int*   __restrict__ mask,
    const float* __restrict__ trans,
    const float* __restrict__ start_t,
    const float* __restrict__ end_t,
    float* __restrict__ tags_out)    // (B, T) as float
{
    const int b = blockIdx.x;
    const int j = threadIdx.x;
    __shared__ float alpha_s[N_];
    __shared__ unsigned char bp[T_][N_];   // row 0 unused; ~10.7 KB

    const float* lg = logits + (size_t)b * T_ * N_;
    const int*   mk = mask + (size_t)b * T_;

    float alpha = -INFINITY;
    float tcol[N_];
    if (j < N_) {
        alpha = start_t[j] + lg[j];
#pragma unroll
        for (int i = 0; i < N_; ++i) tcol[i] = trans[i * N_ + j];
    }

    for (int t = 1; t < T_; ++t) {
        if (j < N_) alpha_s[j] = alpha;
        __syncthreads();
        if (j < N_) {
            float best = -INFINITY;
            int arg = 0;
#pragma unroll
            for (int i = 0; i < N_; ++i) {
                const float v = alpha_s[i] + tcol[i];
                if (v > best) { best = v; arg = i; }   // first-max == argmax
            }
            bp[t][j] = (unsigned char)arg;
            if (mk[t]) alpha = best + lg[(size_t)t * N_ + j];
        }
        __syncthreads();
    }

    if (j < N_) alpha_s[j] = alpha + end_t[j];
    __syncthreads();

    if (j == 0) {
        int tag = 0;
        float best = alpha_s[0];
        for (int i = 1; i < N_; ++i)
            if (alpha_s[i] > best) { best = alpha_s[i]; tag = i; }
        tags_out[(size_t)b * T_ + (T_ - 1)] = (float)tag;
        for (int t = T_ - 2; t >= 0; --t) {
            if (mk[t + 1]) tag = bp[t + 1][tag];
            tags_out[(size_t)b * T_ + t] = (float)tag;
        }
    }
}

// ---------------------------------------------------------------------------
// Kernel 4: loss = -sum_b (joint_b - logZ_b)
// ---------------------------------------------------------------------------
__global__ __launch_bounds__(64) void finalize_loss_kernel(
    const float* __restrict__ partial, float* __restrict__ loss_out)
{
    __shared__ float s[64];
    s[threadIdx.x] = partial[threadIdx.x];
    __syncthreads();
    for (int off = 32; off > 0; off >>= 1) {
        if (threadIdx.x < off) s[threadIdx.x] += s[threadIdx.x + off];
        __syncthreads();
    }
    if (threadIdx.x == 0) loss_out[0] = -s[0];
}

// ---------------------------------------------------------------------------
extern "C" void kernel_launch(void* const* d_in, const int* in_sizes, int n_in,
                              void* d_out, int out_size, void* d_ws, size_t ws_size,
                              hipStream_t stream) {
    const float* mlm_out = (const float*)d_in[0];   // (B, T, D)
    const int*   mask    = (const int*)d_in[1];     // (B, T)
    const int*   gold    = (const int*)d_in[2];     // (B, T)
    const float* W       = (const float*)d_in[3];   // (N_, D)
    const float* bias    = (const float*)d_in[4];   // (N_,)
    const float* trans   = (const float*)d_in[5];   // (N_, N_)
    const float* start_t = (const float*)d_in[6];   // (N_,)
    const float* end_t   = (const float*)d_in[7];   // (N_,)

    float* out      = (float*)d_out;
    float* logits   = out;                                  // B*T*N_ floats
    float* loss     = out + (size_t)B_ * T_ * N_;           // 1 float
    float* tags_out = loss + 1;                             // B*T floats
    float* partial  = (float*)d_ws;                         // 64 floats scratch

    gemm_logits_kernel<<<dim3(256), dim3(256), 0, stream>>>(mlm_out, W, bias, logits);

    crf_forward_kernel<<<dim3(B_), dim3(32), 0, stream>>>(
        logits, mask, gold, trans, start_t, end_t, partial);

    crf_viterbi_kernel<<<dim3(B_), dim3(32), 0, stream>>>(
        logits, mask, trans, start_t, end_t, tags_out);

    finalize_loss_kernel<<<dim3(1), dim3(64), 0, stream>>>(partial, loss);
}